// ComplexLayerNorm_3753801417341
// MI455X (gfx1250) — compile-verified
//
#include <hip/hip_runtime.h>

// Problem constants (match reference setup_inputs).
constexpr int B = 64;
constexpr int C = 128;
constexpr int F = 2048;
constexpr float EPS = 1e-4f;

typedef __attribute__((ext_vector_type(2))) float v2f;
typedef __attribute__((ext_vector_type(8))) float v8f;

// ---------------------------------------------------------------------------
// Kernel 1: per-(b,c) row means over F, computed with V_WMMA_F32_16X16X4_F32.
// D = A(16x4) x ones(4x16) + C  =>  every column of C accumulates the row sums
// of A. One block = 16 rows; wave w covers columns [w*256, w*256+256).
// Exact f32 math (ones matrix is layout-agnostic).
// ---------------------------------------------------------------------------
__global__ __launch_bounds__(256) void rowmean_kernel(
    const float* __restrict__ xr, const float* __restrict__ xi,
    float* __restrict__ mean_r, float* __restrict__ mean_i) {
  const int row0 = blockIdx.x * 16;           // 8192 rows / 16 = 512 blocks
  const int wv   = threadIdx.x >> 5;          // 8 waves
  const int lane = threadIdx.x & 31;
  // A-matrix 16x4 f32 layout: lanes 0-15 -> M=lane, K={0,1}; lanes 16-31 -> K={2,3}
  const int m    = lane & 15;
  const int koff = (lane >> 4) << 1;          // 0 or 2
  const float* pr = xr + (size_t)(row0 + m) * F + koff;
  const float* pi = xi + (size_t)(row0 + m) * F + koff;
  const int cbase = wv * 256;

  __shared__ float lds_r[8][16];
  __shared__ float lds_i[8][16];

#if __has_builtin(__builtin_amdgcn_wmma_f32_16x16x4_f32)
  v8f cr = {}; v8f ci = {};
  const v2f ones = {1.0f, 1.0f};
  #pragma unroll 4
  for (int k0 = 0; k0 < 256; k0 += 4) {
    v2f ar = *(const v2f*)(pr + cbase + k0);
    v2f ai = *(const v2f*)(pi + cbase + k0);
    cr = __builtin_amdgcn_wmma_f32_16x16x4_f32(false, ar, false, ones,
                                               (short)0, cr, false, false);
    ci = __builtin_amdgcn_wmma_f32_16x16x4_f32(false, ai, false, ones,
                                               (short)0, ci, false, false);
  }
  // C/D layout: VGPR r, lanes 0-15 -> M=r; lanes 16-31 -> M=r+8. Column N=0
  // lives in lane 0 (rows 0..7) and lane 16 (rows 8..15).
  if (lane == 0 || lane == 16) {
    const int half = (lane >> 4) * 8;
    #pragma unroll
    for (int r = 0; r < 8; ++r) {
      lds_r[wv][half + r] = cr[r];
      lds_i[wv][half + r] = ci[r];
    }
  }
#else
  // Fallback: scalar accumulate + cross-half shuffle combine.
  float sr = 0.0f, si = 0.0f;
  #pragma unroll 4
  for (int k0 = 0; k0 < 256; k0 += 4) {
    v2f ar = *(const v2f*)(pr + cbase + k0);
    v2f ai = *(const v2f*)(pi + cbase + k0);
    sr += ar.x + ar.y;
    si += ai.x + ai.y;
  }
  sr += __shfl_down(sr, 16);
  si += __shfl_down(si, 16);
  if (lane < 16) { lds_r[wv][lane] = sr; lds_i[wv][lane] = si; }
#endif
  __syncthreads();
  if (threadIdx.x < 32) {
    const int  t    = threadIdx.x & 15;
    const bool imag = threadIdx.x >= 16;
    float s = 0.0f;
    #pragma unroll
    for (int w = 0; w < 8; ++w) s += imag ? lds_i[w][t] : lds_r[w][t];
    const float mval = s * (1.0f / (float)F);
    if (imag) mean_i[row0 + t] = mval;
    else      mean_r[row0 + t] = mval;
  }
}

// ---------------------------------------------------------------------------
// Kernel 2: per-(c,f) covariance contributions over b:
//   d_rr = sum_b xr^2 - (sum_b xr)^2 / B   (and ri, ii)
// Written to a [C][3][F] partial buffer (deterministic; no atomics).
// Second pass over x hits L2 (both tensors = 128 MiB < 192 MB L2).
// ---------------------------------------------------------------------------
__global__ __launch_bounds__(256) void colstat_kernel(
    const float* __restrict__ xr, const float* __restrict__ xi,
    float* __restrict__ partial) {
  const int c = blockIdx.x >> 3;                         // F/256 = 8 tiles per c
  const int f = ((blockIdx.x & 7) << 8) + threadIdx.x;
  const float* pr = xr + (size_t)c * F + f;
  const float* pi = xi + (size_t)c * F + f;
  const size_t stride = (size_t)C * F;                   // 1 MiB between b's
  float tr = 0.f, ti = 0.f, srr = 0.f, sri = 0.f, sii = 0.f;
  #pragma unroll 4
  for (int b = 0; b < B; ++b) {
    if (b + 8 < B) {
      __builtin_prefetch(pr + (size_t)(b + 8) * stride, 0, 1);  // global_prefetch_b8
      __builtin_prefetch(pi + (size_t)(b + 8) * stride, 0, 1);
    }
    const float r  = pr[(size_t)b * stride];
    const float im = pi[(size_t)b * stride];
    tr += r;  ti += im;
    srr = fmaf(r, r, srr);
    sri = fmaf(r, im, sri);
    sii = fmaf(im, im, sii);
  }
  const float invB = 1.0f / (float)B;
  float* base = partial + (size_t)c * 3 * F;
  base[0 * F + f] = srr - tr * tr * invB;
  base[1 * F + f] = sri - tr * ti * invB;
  base[2 * F + f] = sii - ti * ti * invB;
}

// ---------------------------------------------------------------------------
// Kernel 3: per-f deterministic reduce over c, closed-form SPD 2x2 inverse
// sqrt, fold gamma into 2x2 coefficient matrix:
//   out_r = a11*zr + a12*zi + beta_r ; out_i = a21*zr + a22*zi + beta_i
// ---------------------------------------------------------------------------
__global__ __launch_bounds__(256) void finalize_kernel(
    const float* __restrict__ partial,
    const float* __restrict__ gamma_r, const float* __restrict__ gamma_i,
    float4* __restrict__ coef) {
  const int f = blockIdx.x * 256 + threadIdx.x;
  if (f >= F) return;
  float srr = 0.f, sri = 0.f, sii = 0.f;
  for (int c = 0; c < C; ++c) {
    const float* base = partial + (size_t)c * 3 * F;
    srr += base[0 * F + f];
    sri += base[1 * F + f];
    sii += base[2 * F + f];
  }
  const float inv_nm1 = 1.0f / (float)(B * C - 1);
  const float a  = srr * inv_nm1 + EPS;          // M = cov + eps*I
  const float bb = sri * inv_nm1;
  const float d  = sii * inv_nm1 + EPS;
  const float tr  = a + d;
  const float det = a * d - bb * bb;
  const float s   = sqrtf(det);
  const float tau = sqrtf(tr + 2.0f * s);
  const float inv = 1.0f / (s * tau);
  const float wrr = (d + s) * inv;
  const float wii = (a + s) * inv;
  const float wri = -bb * inv;
  const float gr = gamma_r[f], gi = gamma_i[f];
  coef[f] = make_float4(gr * wrr - gi * wri,     // a11
                        gr * wri - gi * wii,     // a12
                        gi * wrr + gr * wri,     // a21
                        gi * wri + gr * wii);    // a22
}

// ---------------------------------------------------------------------------
// Kernel 4: whitening + affine, 2 features per thread, one interleaved
// float4 store per thread (peak-width streaming; output is the HBM cost).
// ---------------------------------------------------------------------------
__global__ __launch_bounds__(256) void apply_kernel(
    const float* __restrict__ xr, const float* __restrict__ xi,
    const float* __restrict__ mean_r, const float* __restrict__ mean_i,
    const float4* __restrict__ coef,
    const float* __restrict__ beta_r, const float* __restrict__ beta_i,
    float4* __restrict__ out) {
  const size_t t   = (size_t)blockIdx.x * 256 + threadIdx.x;   // B*C*F/2 threads
  const int    row = (int)(t >> 10);                           // F/2 = 1024
  const int    f   = ((int)(t & 1023)) << 1;
  const size_t base = (size_t)row * F + f;
  const v2f xrv = *(const v2f*)(xr + base);
  const v2f xiv = *(const v2f*)(xi + base);
  const float mr = mean_r[row];
  const float mi = mean_i[row];
  const float4 c0 = coef[f];
  const float4 c1 = coef[f + 1];
  const v2f br = *(const v2f*)(beta_r + f);
  const v2f bi = *(const v2f*)(beta_i + f);
  const float zr0 = xrv.x - mr, zi0 = xiv.x - mi;
  const float zr1 = xrv.y - mr, zi1 = xiv.y - mi;
  float4 o;
  o.x = fmaf(c0.x, zr0, fmaf(c0.y, zi0, br.x));
  o.y = fmaf(c0.z, zr0, fmaf(c0.w, zi0, bi.x));
  o.z = fmaf(c1.x, zr1, fmaf(c1.y, zi1, br.y));
  o.w = fmaf(c1.z, zr1, fmaf(c1.w, zi1, bi.y));
  out[base >> 1] = o;   // (row*F + f)*2 floats == float4 index base/2
}

// ---------------------------------------------------------------------------
extern "C" void kernel_launch(void* const* d_in, const int* in_sizes, int n_in,
                              void* d_out, int out_size, void* d_ws, size_t ws_size,
                              hipStream_t stream) {
  const float* xr      = (const float*)d_in[0];
  const float* xi      = (const float*)d_in[1];
  const float* gamma_r = (const float*)d_in[2];
  const float* gamma_i = (const float*)d_in[3];
  const float* beta_r  = (const float*)d_in[4];
  const float* beta_i  = (const float*)d_in[5];
  float* out = (float*)d_out;

  // Workspace layout (floats), every word overwritten each launch:
  //   partial : C*3*F   = 786432
  //   mean_r  : B*C     =   8192
  //   mean_i  : B*C     =   8192
  //   coef    : 4*F     =   8192   (float4-aligned: offset 802816*4 bytes)
  float* ws      = (float*)d_ws;
  float* partial = ws;
  float* mean_r  = partial + (size_t)C * 3 * F;
  float* mean_i  = mean_r + B * C;
  float* coef    = mean_i + B * C;

  rowmean_kernel<<<(B * C) / 16, 256, 0, stream>>>(xr, xi, mean_r, mean_i);
  colstat_kernel<<<C * (F / 256), 256, 0, stream>>>(xr, xi, partial);
  finalize_kernel<<<F / 256, 256, 0, stream>>>(partial, gamma_r, gamma_i,
                                               (float4*)coef);
  apply_kernel<<<(B * C * (F / 2)) / 256, 256, 0, stream>>>(
      xr, xi, mean_r, mean_i, (const float4*)coef, beta_r, beta_i,
      (float4*)out);
}